// ContextHyperLinearSSM_90314572300307
// MI455X (gfx1250) — compile-verified
//
#include <hip/hip_runtime.h>

// Problem constants (match reference)
#define NB    4096
#define SDIM  256
#define ADIM  128
#define NCTX  512

#define XPITCH (SDIM + 4)   // pad LDS rows so 16-lane fragment loads spread banks
#define UPITCH (ADIM + 4)

typedef __attribute__((ext_vector_type(2))) float v2f;
typedef __attribute__((ext_vector_type(8))) float v8f;

// ---------------- grouping (count-sort samples by context) ----------------

__global__ void zero_counts_kernel(int* counts) {
    counts[blockIdx.x * blockDim.x + threadIdx.x] = 0;
}

__global__ void hist_kernel(const int* __restrict__ ctx, int* counts) {
    int i = blockIdx.x * blockDim.x + threadIdx.x;
    if (i < NB) atomicAdd(&counts[ctx[i]], 1);
}

__global__ void scan_kernel(const int* __restrict__ counts, int* offsets, int* cursor) {
    if (threadIdx.x == 0 && blockIdx.x == 0) {
        int s = 0;
        for (int c = 0; c < NCTX; ++c) { offsets[c] = s; cursor[c] = s; s += counts[c]; }
    }
}

__global__ void scatter_kernel(const int* __restrict__ ctx, int* cursor, int* order) {
    int i = blockIdx.x * blockDim.x + threadIdx.x;
    if (i < NB) { int p = atomicAdd(&cursor[ctx[i]], 1); order[p] = i; }
}

// ---------------- main grouped-GEMM kernel (f32 WMMA) ----------------
// grid = NCTX blocks, 256 threads = 8 waves.
// Wave w owns TWO 16-column N-tiles: columns [16w,16w+16) and [16w+128,16w+144),
// so each LDS A-fragment load feeds two v_wmma_f32_16x16x4_f32.

__global__ __launch_bounds__(256) void ctx_gemm_kernel(
    const float* __restrict__ x, const float* __restrict__ u,
    const float* __restrict__ WA, const float* __restrict__ WB,
    const unsigned char* __restrict__ adj_xx, const unsigned char* __restrict__ adj_xu,
    const int* __restrict__ offsets, const int* __restrict__ counts,
    const int* __restrict__ order, float* __restrict__ out)
{
    const int c   = blockIdx.x;
    const int cnt = counts[c];
    if (cnt == 0) return;
    const int begin = offsets[c];

    __shared__ float xs[16 * XPITCH];
    __shared__ float us[16 * UPITCH];

    const int tid  = threadIdx.x;
    const int lane = tid & 31;
    const int wave = tid >> 5;
    const int s0   = wave * 16;          // first N-tile base column
    const int s1   = s0 + 128;           // second N-tile base column

    const int half  = lane >> 4;         // 0: lanes 0-15, 1: lanes 16-31
    const int sub   = lane & 15;         // M index (A-frag) / N index (B-frag, store)
    const int khalf = half * 2;          // lanes 16-31 hold K+2,K+3

    const float* WAc = WA + (size_t)c * SDIM * SDIM;
    const float* WBc = WB + (size_t)c * ADIM * SDIM;

    for (int m0 = 0; m0 < cnt; m0 += 16) {
        __syncthreads();  // previous chunk's LDS reads done before overwrite

        // Stage up to 16 gathered x rows and u rows into LDS (zeros for tail).
        for (int t = tid; t < 16 * SDIM; t += 256) {
            int r = t >> 8, k = t & (SDIM - 1);
            float v = 0.0f;
            int m = m0 + r;
            if (m < cnt) v = x[(size_t)order[begin + m] * SDIM + k];
            xs[r * XPITCH + k] = v;
        }
        for (int t = tid; t < 16 * ADIM; t += 256) {
            int r = t >> 7, k = t & (ADIM - 1);
            float v = 0.0f;
            int m = m0 + r;
            if (m < cnt) v = u[(size_t)order[begin + m] * ADIM + k];
            us[r * UPITCH + k] = v;
        }
        __syncthreads();

        v8f acc0 = {};
        v8f acc1 = {};

        // ---- x @ (WA ⊙ adj_xx), K = 256, step 4, two N-tiles per A-fragment ----
        #pragma unroll 4
        for (int k0 = 0; k0 < SDIM; k0 += 4) {
            const int kr = k0 + khalf;
            v2f a;
            a.x = xs[sub * XPITCH + kr];
            a.y = xs[sub * XPITCH + kr + 1];

            const size_t r0 = (size_t)kr * SDIM;
            const size_t r1 = (size_t)(kr + 1) * SDIM;

            v2f b0, b1;
            b0.x = WAc[r0 + s0 + sub] * (float)adj_xx[r0 + s0 + sub];
            b0.y = WAc[r1 + s0 + sub] * (float)adj_xx[r1 + s0 + sub];
            b1.x = WAc[r0 + s1 + sub] * (float)adj_xx[r0 + s1 + sub];
            b1.y = WAc[r1 + s1 + sub] * (float)adj_xx[r1 + s1 + sub];

            if ((k0 & 15) == 0) {
                __builtin_prefetch(&WAc[r0 + 16 * SDIM + s0 + sub], 0, 0);
                __builtin_prefetch(&WAc[r0 + 16 * SDIM + s1 + sub], 0, 0);
            }

            acc0 = __builtin_amdgcn_wmma_f32_16x16x4_f32(
                false, a, false, b0, (short)0, acc0, false, false);
            acc1 = __builtin_amdgcn_wmma_f32_16x16x4_f32(
                false, a, false, b1, (short)0, acc1, false, false);
        }

        // ---- u @ (WB ⊙ adj_xu), K = 128, step 4, two N-tiles per A-fragment ----
        #pragma unroll 4
        for (int k0 = 0; k0 < ADIM; k0 += 4) {
            const int kr = k0 + khalf;
            v2f a;
            a.x = us[sub * UPITCH + kr];
            a.y = us[sub * UPITCH + kr + 1];

            const size_t r0 = (size_t)kr * SDIM;
            const size_t r1 = (size_t)(kr + 1) * SDIM;

            v2f b0, b1;
            b0.x = WBc[r0 + s0 + sub] * (float)adj_xu[r0 + s0 + sub];
            b0.y = WBc[r1 + s0 + sub] * (float)adj_xu[r1 + s0 + sub];
            b1.x = WBc[r0 + s1 + sub] * (float)adj_xu[r0 + s1 + sub];
            b1.y = WBc[r1 + s1 + sub] * (float)adj_xu[r1 + s1 + sub];

            if ((k0 & 15) == 0) {
                __builtin_prefetch(&WBc[r0 + 16 * SDIM + s0 + sub], 0, 0);
                __builtin_prefetch(&WBc[r0 + 16 * SDIM + s1 + sub], 0, 0);
            }

            acc0 = __builtin_amdgcn_wmma_f32_16x16x4_f32(
                false, a, false, b0, (short)0, acc0, false, false);
            acc1 = __builtin_amdgcn_wmma_f32_16x16x4_f32(
                false, a, false, b1, (short)0, acc1, false, false);
        }

        // ---- scatter C/D: VGPR j -> M=j (lanes 0-15) / M=8+j (lanes 16-31), N=sub ----
        #pragma unroll
        for (int j = 0; j < 8; ++j) {
            int m = m0 + half * 8 + j;
            if (m < cnt) {
                int b = order[begin + m];
                out[(size_t)b * SDIM + s0 + sub] = acc0[j];
                out[(size_t)b * SDIM + s1 + sub] = acc1[j];
            }
        }
    }
}

// ---------------- host-side launcher ----------------

extern "C" void kernel_launch(void* const* d_in, const int* in_sizes, int n_in,
                              void* d_out, int out_size, void* d_ws, size_t ws_size,
                              hipStream_t stream) {
    const float*         x      = (const float*)d_in[0];
    const float*         u      = (const float*)d_in[1];
    const float*         WA     = (const float*)d_in[2];
    const float*         WB     = (const float*)d_in[3];
    const unsigned char* adj_xx = (const unsigned char*)d_in[4];
    const unsigned char* adj_xu = (const unsigned char*)d_in[5];
    const int*           ctx    = (const int*)d_in[6];
    float*               out    = (float*)d_out;

    int* ws      = (int*)d_ws;
    int* counts  = ws;               // [NCTX]
    int* offsets = ws + NCTX;        // [NCTX]
    int* cursor  = ws + 2 * NCTX;    // [NCTX]
    int* order   = ws + 3 * NCTX;    // [NB]

    zero_counts_kernel<<<NCTX / 256, 256, 0, stream>>>(counts);
    hist_kernel<<<NB / 256, 256, 0, stream>>>(ctx, counts);
    scan_kernel<<<1, 32, 0, stream>>>(counts, offsets, cursor);
    scatter_kernel<<<NB / 256, 256, 0, stream>>>(ctx, cursor, order);
    ctx_gemm_kernel<<<NCTX, 256, 0, stream>>>(x, u, WA, WB, adj_xx, adj_xu,
                                              offsets, counts, order, out);
}